// Conv1d1x1_11871289606702
// MI455X (gfx1250) — compile-verified
//
#include <hip/hip_runtime.h>

// ---------------- problem constants ----------------
#define BTOK   8192            // unique tokens
#define DIM    512             // d_model
#define NCOUT  512
#define NEXP   16
#define TOPK   2
#define NROUTE (BTOK*TOPK)     // 16384 routed rows
#define MAXTILES (NROUTE/16 + NEXP)   // 1040 worst-case 16-row tiles

// ---------------- workspace layout (bytes) ----------------
#define WS_COUNTS   0u                      // 16 int
#define WS_CURSOR   64u                     // 16 int
#define WS_OFFS     128u                    // 17 int
#define WS_TILEOFF  256u                    // 17 int
#define WS_TOTTILES 384u                    // 1 int
#define WS_GIDX     1024u                   // NROUTE int
#define WS_GW       (WS_GIDX + NROUTE*4u)   // NROUTE float
#define WS_PTOK     (WS_GW   + NROUTE*4u)   // NROUTE int
#define WS_PW       (WS_PTOK + NROUTE*4u)   // NROUTE float
#define WS_PDST     (WS_PW   + NROUTE*4u)   // NROUTE int
#define WS_EOUT     (1u<<19)                // NROUTE*NCOUT floats = 32 MB

typedef float v2f __attribute__((ext_vector_type(2)));
typedef float v4f __attribute__((ext_vector_type(4)));
typedef float v8f __attribute__((ext_vector_type(8)));
typedef int   v4i __attribute__((ext_vector_type(4)));
typedef v4i __attribute__((address_space(1))) gas_v4i;   // global
typedef v4i __attribute__((address_space(3))) las_v4i;   // LDS

// ---------------- init ----------------
__global__ void init_kernel(int* counts) {
    if (threadIdx.x < NEXP) counts[threadIdx.x] = 0;
}

// ---------------- gate: logits -> top2 -> softmax -> histogram ----------------
// one wave32 per token; 8 tokens per 256-thread block; Wg staged in LDS (32 KB)
__global__ void gate_kernel(const float* __restrict__ x,
                            const float* __restrict__ Wg,
                            const float* __restrict__ bg,
                            int* __restrict__ counts,
                            int* __restrict__ gidx,
                            float* __restrict__ gw) {
    __shared__ float sWg[DIM * NEXP];
    for (int i = threadIdx.x; i < DIM * NEXP; i += 256) sWg[i] = Wg[i];
    __syncthreads();

    const int wave = threadIdx.x >> 5;
    const int lane = threadIdx.x & 31;
    const int t = blockIdx.x * 8 + wave;
    const float* tok = x + (size_t)t * DIM;

    float acc[NEXP];
#pragma unroll
    for (int e = 0; e < NEXP; ++e) acc[e] = 0.f;

#pragma unroll 4
    for (int i = 0; i < DIM / 32; ++i) {
        const int d = lane + i * 32;
        const float xv = tok[d];
        const float* wrow = &sWg[d * NEXP];
#pragma unroll
        for (int e = 0; e < NEXP; ++e) acc[e] = fmaf(xv, wrow[e], acc[e]);
    }
#pragma unroll
    for (int e = 0; e < NEXP; ++e) {
        acc[e] += __shfl_xor(acc[e], 16, 32);
        acc[e] += __shfl_xor(acc[e], 8, 32);
        acc[e] += __shfl_xor(acc[e], 4, 32);
        acc[e] += __shfl_xor(acc[e], 2, 32);
        acc[e] += __shfl_xor(acc[e], 1, 32);
    }
    if (lane == 0) {
        float lg[NEXP];
#pragma unroll
        for (int e = 0; e < NEXP; ++e) lg[e] = acc[e] + bg[e];
        // top-2, lowest index wins ties (matches top_k)
        int i0 = 0; float v0 = lg[0];
        for (int e = 1; e < NEXP; ++e) if (lg[e] > v0) { v0 = lg[e]; i0 = e; }
        int i1 = (i0 == 0) ? 1 : 0; float v1 = lg[i1];
        for (int e = 0; e < NEXP; ++e)
            if (e != i0 && lg[e] > v1) { v1 = lg[e]; i1 = e; }
        // softmax over the two selected logits (v0 >= v1)
        const float e0 = 1.0f, e1 = __expf(v1 - v0);
        const float inv = 1.0f / (e0 + e1);
        gidx[t * 2 + 0] = i0; gidx[t * 2 + 1] = i1;
        gw[t * 2 + 0] = e0 * inv; gw[t * 2 + 1] = e1 * inv;
        atomicAdd(&counts[i0], 1);
        atomicAdd(&counts[i1], 1);
    }
}

// ---------------- scan: offsets + tile offsets ----------------
__global__ void scan_kernel(const int* __restrict__ counts,
                            int* __restrict__ offs,
                            int* __restrict__ tileoff,
                            int* __restrict__ tottiles,
                            int* __restrict__ cursor) {
    if (threadIdx.x == 0) {
        int o = 0, to = 0;
        for (int e = 0; e < NEXP; ++e) {
            offs[e] = o; tileoff[e] = to;
            o += counts[e];
            to += (counts[e] + 15) >> 4;
        }
        offs[NEXP] = o; tileoff[NEXP] = to; *tottiles = to;
    }
    if (threadIdx.x < NEXP) cursor[threadIdx.x] = 0;
}

// ---------------- scatter: compact routed rows per expert ----------------
__global__ void scatter_kernel(const int* __restrict__ gidx,
                               const float* __restrict__ gw,
                               const int* __restrict__ offs,
                               int* __restrict__ cursor,
                               int* __restrict__ ptok,
                               float* __restrict__ pw,
                               int* __restrict__ pdst) {
    const int t = blockIdx.x * 256 + threadIdx.x;
#pragma unroll
    for (int k = 0; k < TOPK; ++k) {
        const int e = gidx[t * 2 + k];
        const int pos = atomicAdd(&cursor[e], 1);
        const int idx = offs[e] + pos;
        ptok[idx] = t;
        pw[idx]   = gw[t * 2 + k];
        pdst[idx] = t * 2 + k;       // unique output slot -> deterministic
    }
}

// ---------------- routed expert GEMM with V_WMMA_F32_16X16X4_F32 ----------------
// block = 128 (4 waves); each wave owns one 16x16 C tile; block covers 16 rows x 64 cols.
// A (16x512 gathered tokens) staged in LDS with row stride 516 to avoid bank conflicts.
#define LDA 516
__global__ void expert_gemm_kernel(const float* __restrict__ x,
                                   const float* __restrict__ We,
                                   const float* __restrict__ be,
                                   const int* __restrict__ offs,
                                   const int* __restrict__ tileoff,
                                   const int* __restrict__ tottiles,
                                   const int* __restrict__ ptok,
                                   const float* __restrict__ pw,
                                   const int* __restrict__ pdst,
                                   float* __restrict__ eout) {
    __shared__ float sA[16 * LDA];
    __shared__ float sW[16];
    __shared__ int   sTok[16];
    __shared__ int   sDst[16];
    __shared__ int   sInfo[3];   // expert, row0, nrows

    if (blockIdx.x >= *tottiles) return;   // uniform per block

    const int tid = threadIdx.x;
    if (tid == 0) {
        const int tile = blockIdx.x;
        int e = 0;
        while (e < NEXP - 1 && tile >= tileoff[e + 1]) ++e;
        const int row0 = offs[e] + (tile - tileoff[e]) * 16;
        int nrows = offs[e + 1] - row0;
        if (nrows > 16) nrows = 16;
        sInfo[0] = e; sInfo[1] = row0; sInfo[2] = nrows;
    }
    __syncthreads();
    const int e     = sInfo[0];
    const int row0  = sInfo[1];
    const int nrows = sInfo[2];

    if (tid < 16) {
        if (tid < nrows) {
            const int r = row0 + tid;
            sTok[tid] = ptok[r]; sW[tid] = pw[r]; sDst[tid] = pdst[r];
        } else {
            sTok[tid] = 0; sW[tid] = 0.f; sDst[tid] = -1;
        }
    }
    __syncthreads();

    // gather A tile: 16 rows x 512 floats, 16B chunks, 16 per thread.
    // CDNA5 path: GLOBAL_LOAD_ASYNC_TO_LDS_B128 (ASYNCcnt-tracked, bypasses VGPRs).
    for (int i = tid; i < 16 * 128; i += 128) {
        const int row = i >> 7, c4 = i & 127;
        const float* src = x + (size_t)sTok[row] * DIM + c4 * 4;
        float* dst = &sA[row * LDA + c4 * 4];
#if __has_builtin(__builtin_amdgcn_global_load_async_to_lds_b128)
        __builtin_amdgcn_global_load_async_to_lds_b128(
            (gas_v4i*)(uintptr_t)src,
            (las_v4i*)(unsigned)(uintptr_t)dst,   // LDS byte offset (addr[31:0])
            0, 0);
#else
        const unsigned ldsOff = (unsigned)(uintptr_t)dst;  // LDS byte address
        asm volatile("global_load_async_to_lds_b128 %0, %1, off"
                     :: "v"(ldsOff), "v"(src) : "memory");
#endif
    }
    // each wave drains its own async copies, then block barrier publishes LDS
#if __has_builtin(__builtin_amdgcn_s_wait_asynccnt)
    __builtin_amdgcn_s_wait_asynccnt(0);
#else
    asm volatile("s_wait_asynccnt 0x0" ::: "memory");
#endif
    __syncthreads();

    const int wv = tid >> 5, lane = tid & 31;
    const int c0 = blockIdx.y * 64 + wv * 16;
    const int n  = lane & 15;             // N for B/C/D, M for A
    const int kh = (lane >> 4) * 2;       // K half-select (0 or 2)

    // B operand: B[k][n] = We[e][c0+n][k]; lane holds {k0+kh, k0+kh+1} as float2
    const float* pB = We + ((size_t)e * NCOUT + c0 + n) * DIM + kh;
    // A operand from LDS: A[m][k], lane m = lane&15, same K split
    const float* pA = &sA[(lane & 15) * LDA + kh];

    // fold expert bias into the accumulator (bias depends only on N)
    const float bias = be[e * NCOUT + c0 + n];
    v8f acc;
#pragma unroll
    for (int r = 0; r < 8; ++r) acc[r] = bias;

#pragma unroll 8
    for (int k0 = 0; k0 < DIM; k0 += 4) {
        v2f a, b;
        a[0] = pA[k0]; a[1] = pA[k0 + 1];
        const float2 bv = *(const float2*)(pB + k0);
        b[0] = bv.x; b[1] = bv.y;
        acc = __builtin_amdgcn_wmma_f32_16x16x4_f32(
            false, a, false, b, (short)0, acc, false, false);
    }

    // D layout: VGPR r, lane L -> (M = r + 8*(L>>4), N = L&15); scale row by gate weight
    if (nrows == 16) {
        // uniform fast path: no exec-mask dance (vast majority of tiles)
#pragma unroll
        for (int r = 0; r < 8; ++r) {
            const int M = r + 8 * (lane >> 4);
            eout[(size_t)sDst[M] * NCOUT + c0 + n] = acc[r] * sW[M];
        }
    } else {
#pragma unroll
        for (int r = 0; r < 8; ++r) {
            const int M = r + 8 * (lane >> 4);
            if (M < nrows) {
                eout[(size_t)sDst[M] * NCOUT + c0 + n] = acc[r] * sW[M];
            }
        }
    }
}

// ---------------- combine + 8x replication (pure bandwidth, NT stores) ----------------
__global__ void combine_kernel(const float* __restrict__ eout,
                               float* __restrict__ out) {
    const int gid = blockIdx.x * 256 + threadIdx.x;  // B*128 threads, float4 each
    const int m  = gid >> 7;
    const int c4 = gid & 127;
    const v4f* ev = (const v4f*)eout;
    const v4f a = __builtin_nontemporal_load(&ev[(size_t)(m * 2 + 0) * 128 + c4]);
    const v4f b = __builtin_nontemporal_load(&ev[(size_t)(m * 2 + 1) * 128 + c4]);
    const v4f v = a + b;
    v4f* ov = (v4f*)out;
#pragma unroll
    for (int r = 0; r < 8; ++r) {
        __builtin_nontemporal_store(v, &ov[((size_t)m + (size_t)r * BTOK) * 128 + c4]);
    }
}

// ---------------- launch ----------------
extern "C" void kernel_launch(void* const* d_in, const int* in_sizes, int n_in,
                              void* d_out, int out_size, void* d_ws, size_t ws_size,
                              hipStream_t stream) {
    const float* x  = (const float*)d_in[0];
    const float* Wg = (const float*)d_in[1];
    const float* bg = (const float*)d_in[2];
    const float* We = (const float*)d_in[3];
    const float* be = (const float*)d_in[4];

    char* ws = (char*)d_ws;
    int*   counts   = (int*)(ws + WS_COUNTS);
    int*   cursor   = (int*)(ws + WS_CURSOR);
    int*   offs     = (int*)(ws + WS_OFFS);
    int*   tileoff  = (int*)(ws + WS_TILEOFF);
    int*   tottiles = (int*)(ws + WS_TOTTILES);
    int*   gidx     = (int*)(ws + WS_GIDX);
    float* gw       = (float*)(ws + WS_GW);
    int*   ptok     = (int*)(ws + WS_PTOK);
    float* pw       = (float*)(ws + WS_PW);
    int*   pdst     = (int*)(ws + WS_PDST);
    float* eout     = (float*)(ws + WS_EOUT);

    init_kernel<<<1, 32, 0, stream>>>(counts);
    gate_kernel<<<BTOK / 8, 256, 0, stream>>>(x, Wg, bg, counts, gidx, gw);
    scan_kernel<<<1, 32, 0, stream>>>(counts, offs, tileoff, tottiles, cursor);
    scatter_kernel<<<BTOK / 256, 256, 0, stream>>>(gidx, gw, offs, cursor, ptok, pw, pdst);
    dim3 gg(MAXTILES, NCOUT / 64);
    expert_gemm_kernel<<<gg, 128, 0, stream>>>(x, We, be, offs, tileoff, tottiles,
                                               ptok, pw, pdst, eout);
    combine_kernel<<<(BTOK * 128) / 256, 256, 0, stream>>>(eout, (float*)d_out);
}